// WeightedKabschAlgorithm_1451698946368
// MI455X (gfx1250) — compile-verified
//
#include <hip/hip_runtime.h>
#include <hip/hip_bf16.h>
#include <math.h>

#define NACC 32
#define NBLK 1024
#define TPB  256

typedef __attribute__((ext_vector_type(2))) float v2f;
typedef __attribute__((ext_vector_type(8))) float v8f;

// ---------------------------------------------------------------------------
// Accumulate one point into the 32 running sums.
//  [0] count(w>0)  [1] sum w
//  [2..4]  sum w*x          [5..7]  sum w*y
//  [8..16] sum w*y_o*x_i    (o*3+i)
//  [17..19] sum x           [20..22] sum y
//  [23..31] sum y_o*x_i     (plain, for the +EPS branch)
// ---------------------------------------------------------------------------
__device__ __forceinline__ void accum_pt(float* a, float w,
                                         float x0, float x1, float x2,
                                         float y0, float y1, float y2) {
  a[0] += (w > 0.0f) ? 1.0f : 0.0f;
  a[1] += w;
  a[2] += w * x0; a[3] += w * x1; a[4] += w * x2;
  float wy0 = w * y0, wy1 = w * y1, wy2 = w * y2;
  a[5] += wy0; a[6] += wy1; a[7] += wy2;
  a[8]  += wy0 * x0; a[9]  += wy0 * x1; a[10] += wy0 * x2;
  a[11] += wy1 * x0; a[12] += wy1 * x1; a[13] += wy1 * x2;
  a[14] += wy2 * x0; a[15] += wy2 * x1; a[16] += wy2 * x2;
  a[17] += x0; a[18] += x1; a[19] += x2;
  a[20] += y0; a[21] += y1; a[22] += y2;
  a[23] += y0 * x0; a[24] += y0 * x1; a[25] += y0 * x2;
  a[26] += y1 * x0; a[27] += y1 * x1; a[28] += y1 * x2;
  a[29] += y2 * x0; a[30] += y2 * x1; a[31] += y2 * x2;
}

// ---------------------------------------------------------------------------
// Phase 1: bandwidth-bound streaming reduction. Each thread owns 4 consecutive
// points per step (3x float4 per cloud + 1x float4 of weights = 7 b128 loads).
// Wave32 butterfly reduce, then per-block partials -> d_ws. Fully deterministic
// (fixed partitioning, no atomics).
// ---------------------------------------------------------------------------
__global__ __launch_bounds__(TPB)
void kabsch_reduce(const float* __restrict__ c0,
                   const float* __restrict__ c1,
                   const float* __restrict__ wts,
                   float* __restrict__ partials, int n) {
  float acc[NACC];
#pragma unroll
  for (int i = 0; i < NACC; ++i) acc[i] = 0.0f;

  const int ngroups = n >> 2;  // groups of 4 points
  const int gstart  = blockIdx.x * blockDim.x + threadIdx.x;
  const int gstride = gridDim.x * blockDim.x;

  for (int g = gstart; g < ngroups; g += gstride) {
    const float4* p0 = (const float4*)(c0 + 12ull * (unsigned long long)g);
    const float4* p1 = (const float4*)(c1 + 12ull * (unsigned long long)g);
    float4 a0 = p0[0], a1 = p0[1], a2 = p0[2];
    float4 b0 = p1[0], b1 = p1[1], b2 = p1[2];
    float4 w4 = *(const float4*)(wts + 4ull * (unsigned long long)g);
    accum_pt(acc, w4.x, a0.x, a0.y, a0.z, b0.x, b0.y, b0.z);
    accum_pt(acc, w4.y, a0.w, a1.x, a1.y, b0.w, b1.x, b1.y);
    accum_pt(acc, w4.z, a1.z, a1.w, a2.x, b1.z, b1.w, b2.x);
    accum_pt(acc, w4.w, a2.y, a2.z, a2.w, b2.y, b2.z, b2.w);
  }
  // scalar tail (N % 4), same grid-stride partition -> deterministic
  for (int i = (ngroups << 2) + gstart; i < n; i += gstride) {
    accum_pt(acc, wts[i],
             c0[3 * i + 0], c0[3 * i + 1], c0[3 * i + 2],
             c1[3 * i + 0], c1[3 * i + 1], c1[3 * i + 2]);
  }

  // wave32 butterfly reduction
#pragma unroll
  for (int i = 0; i < NACC; ++i) {
    float v = acc[i];
#pragma unroll
    for (int off = 16; off > 0; off >>= 1) v += __shfl_xor(v, off, 32);
    acc[i] = v;
  }

  __shared__ float red[TPB / 32][NACC];
  const int lane = threadIdx.x & 31;
  const int wid  = threadIdx.x >> 5;
  if (lane == 0) {
#pragma unroll
    for (int i = 0; i < NACC; ++i) red[wid][i] = acc[i];
  }
  __syncthreads();
  if (threadIdx.x < NACC) {
    float s = 0.0f;
#pragma unroll
    for (int wv = 0; wv < TPB / 32; ++wv) s += red[wv][threadIdx.x];
    partials[(unsigned long long)blockIdx.x * NACC + threadIdx.x] = s;
  }
}

__device__ __forceinline__ float det3(const float* M) {
  return M[0] * (M[4] * M[8] - M[5] * M[7])
       - M[1] * (M[3] * M[8] - M[5] * M[6])
       + M[2] * (M[3] * M[7] - M[4] * M[6]);
}

// ---------------------------------------------------------------------------
// Phase 2: one wave32. Reduce the (NBLK x 32) partials matrix with
// V_WMMA_F32_16X16X4_F32: A = ones(16x4) so D = A*B + C accumulates exact-f32
// column sums of each 4x16 B tile. Then lane 0 runs the 3x3 Jacobi SVD and
// assembles the 4x4 transform.
// ---------------------------------------------------------------------------
__global__ __launch_bounds__(32)
void kabsch_finalize(const float* __restrict__ partials,
                     float* __restrict__ out, int n, int nblk) {
  const int lane = threadIdx.x;
  const int col  = lane & 15;
  const int rofs = (lane >> 4) << 1;  // K slot offset per half-wave

  v2f ones; ones.x = 1.0f; ones.y = 1.0f;
  v8f acc0 = {};
  v8f acc1 = {};

  for (int r = 0; r < nblk; r += 4) {
    const int base = r + rofs;
    v2f b0, b1;
    b0.x = partials[(base + 0) * NACC + col];
    b0.y = partials[(base + 1) * NACC + col];
    b1.x = partials[(base + 0) * NACC + 16 + col];
    b1.y = partials[(base + 1) * NACC + 16 + col];
    acc0 = __builtin_amdgcn_wmma_f32_16x16x4_f32(false, ones, false, b0,
                                                 (short)0, acc0, false, false);
    acc1 = __builtin_amdgcn_wmma_f32_16x16x4_f32(false, ones, false, b1,
                                                 (short)0, acc1, false, false);
  }

  __shared__ float s[NACC];
  if (lane < 16) {          // D row 0: VGPR0, lanes 0..15 hold columns 0..15
    s[lane]      = acc0[0];
    s[lane + 16] = acc1[0];
  }
  __syncthreads();

  if (lane != 0) return;

  const float EPSF = 1.1920929e-07f;
  float cnt = s[0], sw = s[1];
  int not_enough = (cnt < 3.0f);
  float e   = not_enough ? EPSF : 0.0f;
  float cum = sw + e * (float)n;

  float mx[3], my[3], S[9];
  for (int i = 0; i < 3; ++i) mx[i] = (s[2 + i] + e * s[17 + i]) / cum;
  for (int o = 0; o < 3; ++o) my[o] = (s[5 + o] + e * s[20 + o]) / cum;
  for (int o = 0; o < 3; ++o)
    for (int i = 0; i < 3; ++i)
      S[o * 3 + i] = (s[8 + o * 3 + i] + e * s[23 + o * 3 + i]) / cum - my[o] * mx[i];

  // ---- one-sided Jacobi SVD of the 3x3 S:  S = U * diag(d) * V^T ----
  float A[9], V[9] = {1, 0, 0, 0, 1, 0, 0, 0, 1};
  for (int i = 0; i < 9; ++i) A[i] = S[i];
  const int pr[3] = {0, 0, 1}, qr[3] = {1, 2, 2};
  for (int sweep = 0; sweep < 16; ++sweep) {
    for (int k = 0; k < 3; ++k) {
      int p = pr[k], q = qr[k];
      float al = 0.f, be = 0.f, ga = 0.f;
      for (int r = 0; r < 3; ++r) {
        float ap = A[r * 3 + p], aq = A[r * 3 + q];
        al += ap * ap; be += aq * aq; ga += ap * aq;
      }
      if (fabsf(ga) > 1e-30f) {
        float zeta = (be - al) / (2.0f * ga);
        float t = copysignf(1.0f, zeta) / (fabsf(zeta) + sqrtf(1.0f + zeta * zeta));
        float c = 1.0f / sqrtf(1.0f + t * t);
        float sn = c * t;
        for (int r = 0; r < 3; ++r) {
          float ap = A[r * 3 + p], aq = A[r * 3 + q];
          A[r * 3 + p] = c * ap - sn * aq;
          A[r * 3 + q] = sn * ap + c * aq;
          float vp = V[r * 3 + p], vq = V[r * 3 + q];
          V[r * 3 + p] = c * vp - sn * vq;
          V[r * 3 + q] = sn * vp + c * vq;
        }
      }
    }
  }
  float d[3];
  for (int j = 0; j < 3; ++j) {
    float t = 0.f;
    for (int r = 0; r < 3; ++r) t += A[r * 3 + j] * A[r * 3 + j];
    d[j] = sqrtf(t);
  }
  // sort descending, permuting columns of A and V together
  for (int a = 0; a < 2; ++a)
    for (int b = a + 1; b < 3; ++b)
      if (d[b] > d[a]) {
        float td = d[a]; d[a] = d[b]; d[b] = td;
        for (int r = 0; r < 3; ++r) {
          float t1 = A[r * 3 + a]; A[r * 3 + a] = A[r * 3 + b]; A[r * 3 + b] = t1;
          float t2 = V[r * 3 + a]; V[r * 3 + a] = V[r * 3 + b]; V[r * 3 + b] = t2;
        }
      }
  float U[9];
  float dtiny = d[0] * 1e-6f + 1e-30f;
  // u0
  if (d[0] > 1e-30f) for (int r = 0; r < 3; ++r) U[r * 3 + 0] = A[r * 3 + 0] / d[0];
  else { U[0] = 1.f; U[3] = 0.f; U[6] = 0.f; }
  // u1
  if (d[1] > dtiny) for (int r = 0; r < 3; ++r) U[r * 3 + 1] = A[r * 3 + 1] / d[1];
  else {
    float ex = (fabsf(U[0]) < 0.9f) ? 1.f : 0.f, ey = 1.f - ex;
    float cx = U[3] * 0.f - U[6] * ey, cy = U[6] * ex - U[0] * 0.f,
          cz = U[0] * ey - U[3] * ex;
    float nn = sqrtf(cx * cx + cy * cy + cz * cz) + 1e-30f;
    U[1] = cx / nn; U[4] = cy / nn; U[7] = cz / nn;
  }
  // u2
  if (d[2] > dtiny) for (int r = 0; r < 3; ++r) U[r * 3 + 2] = A[r * 3 + 2] / d[2];
  else {
    U[2] = U[3] * U[7] - U[6] * U[4];
    U[5] = U[6] * U[1] - U[0] * U[7];
    U[8] = U[0] * U[4] - U[3] * U[1];
  }

  float tol  = d[0] * 3.0f * EPSF;
  int   rank = (d[0] > tol) + (d[1] > tol) + (d[2] > tol);
  float detS = det3(S);
  float dmul = det3(U) * det3(V);
  float sign_full = (detS < 0.0f) ? -1.0f : 1.0f;
  float sign_def  = (fabsf(dmul + 1.0f) < 1e-4f) ? -1.0f : 1.0f;
  float sf = (rank > 2) ? sign_full : sign_def;

  // R = U * diag(1,1,sf) * V^T ;  t = my - R*mx
  float R[9];
  for (int o = 0; o < 3; ++o)
    for (int i = 0; i < 3; ++i)
      R[o * 3 + i] = U[o * 3 + 0] * V[i * 3 + 0]
                   + U[o * 3 + 1] * V[i * 3 + 1]
                   + sf * U[o * 3 + 2] * V[i * 3 + 2];
  float tv[3];
  for (int o = 0; o < 3; ++o)
    tv[o] = my[o] - (R[o * 3 + 0] * mx[0] + R[o * 3 + 1] * mx[1] + R[o * 3 + 2] * mx[2]);

  for (int o = 0; o < 3; ++o) {
    out[o * 4 + 0] = R[o * 3 + 0];
    out[o * 4 + 1] = R[o * 3 + 1];
    out[o * 4 + 2] = R[o * 3 + 2];
    out[o * 4 + 3] = tv[o];
  }
  out[12] = 0.f; out[13] = 0.f; out[14] = 0.f; out[15] = 1.f;
  out[16] = not_enough ? 1.0f : 0.0f;
}

extern "C" void kernel_launch(void* const* d_in, const int* in_sizes, int n_in,
                              void* d_out, int out_size, void* d_ws, size_t ws_size,
                              hipStream_t stream) {
  const float* c0  = (const float*)d_in[0];   // cloud_t0 (1,N,3)
  const float* c1  = (const float*)d_in[1];   // cloud_t1 (1,N,3)
  const float* wts = (const float*)d_in[2];   // weights  (1,N)
  const int n = in_sizes[2];                  // N
  float* partials = (float*)d_ws;             // NBLK * NACC floats = 128 KB

  kabsch_reduce<<<NBLK, TPB, 0, stream>>>(c0, c1, wts, partials, n);
  kabsch_finalize<<<1, 32, 0, stream>>>(partials, (float*)d_out, n, NBLK);
}